// DynamicGraphLearner_55825984913534
// MI455X (gfx1250) — compile-verified
//
#include <hip/hip_runtime.h>
#include <hip/hip_bf16.h>
#include <math.h>

typedef __attribute__((ext_vector_type(2))) float v2f;
typedef __attribute__((ext_vector_type(8))) float v8f;

#define H 128
#define NEG_SLOPE 0.2f
#define TOPK 16

// ---------------------------------------------------------------------------
// Kernel 0: b1 = W^T a1, b2 = W^T a2; build B matrix (H x 16) col0=b1 col1=b2.
// ---------------------------------------------------------------------------
__global__ void build_b_kernel(const float* __restrict__ W,
                               const float* __restrict__ a,
                               float* __restrict__ B) {
    const int c = threadIdx.x;              // 0..127
    float acc1 = 0.f, acc2 = 0.f;
    #pragma unroll 8
    for (int r = 0; r < H; ++r) {
        const float w = W[r * H + c];       // W[r][c]
        acc1 = fmaf(w, a[r], acc1);
        acc2 = fmaf(w, a[H + r], acc2);
    }
    #pragma unroll
    for (int n = 0; n < 16; ++n) B[c * 16 + n] = 0.f;
    B[c * 16 + 0] = acc1;
    B[c * 16 + 1] = acc2;
}

// ---------------------------------------------------------------------------
// Kernel 1: S = E (N x 128) * B (128 x 16) via V_WMMA_F32_16X16X4_F32.
// One wave32 per 16-row tile. Only output columns 0 (s1) and 1 (s2) are kept.
// A frag (16x4 f32): lane l<16 holds A[l][k],A[l][k+1]; lane l>=16 holds K+2,K+3.
// B frag (4x16 f32): lane l<16 holds B[k][l],B[k+1][l]; lanes 16-31 hold K+2,K+3.
// C (16x16 f32): VGPR v, lanes0-15 -> M=v, lanes16-31 -> M=v+8; N = lane&15.
// ---------------------------------------------------------------------------
__global__ void gemv_wmma_kernel(const float* __restrict__ E,
                                 const float* __restrict__ B,
                                 float* __restrict__ s1,
                                 float* __restrict__ s2) {
    const int tile = blockIdx.x;                 // N/16 tiles
    const int lane = threadIdx.x;                // 0..31
    const int mrow = (lane & 15);
    const int hi   = (lane >> 4);                // 0 or 1
    const int row  = tile * 16 + mrow;
    const float* erow = E + (size_t)row * H;

    v8f acc = {};
    #pragma unroll
    for (int k = 0; k < H; k += 4) {
        const int ka = k + (hi << 1);
        v2f afrag;
        afrag.x = erow[ka + 0];
        afrag.y = erow[ka + 1];
        v2f bfrag;
        bfrag.x = B[(ka + 0) * 16 + mrow];
        bfrag.y = B[(ka + 1) * 16 + mrow];
        acc = __builtin_amdgcn_wmma_f32_16x16x4_f32(
            /*neg_a=*/false, afrag, /*neg_b=*/false, bfrag,
            /*c_mod=*/(short)0, acc, /*reuse_a=*/false, /*reuse_b=*/false);
    }

    // Extract columns N=0 (s1) and N=1 (s2).
    const int mbase = tile * 16 + hi * 8;        // rows mbase..mbase+7 in acc[0..7]
    if (mrow == 0) {
        #pragma unroll
        for (int v = 0; v < 8; ++v) s1[mbase + v] = acc[v];
    } else if (mrow == 1) {
        #pragma unroll
        for (int v = 0; v < 8; ++v) s2[mbase + v] = acc[v];
    }
}

// ---------------------------------------------------------------------------
// Kernel 2: global top-16 of s2 with jax.lax.top_k semantics
// (descending values; ties -> lowest index first). Single block, deterministic.
// ---------------------------------------------------------------------------
__device__ __forceinline__ bool tk_better(float v, int i, float v2, int i2) {
    return (v > v2) || (v == v2 && i < i2);
}

__global__ void topk_kernel(const float* __restrict__ s2,
                            float* __restrict__ tv_out,
                            int* __restrict__ ti_out,
                            int n) {
    __shared__ float sv[64 * TOPK];
    __shared__ int   si[64 * TOPK];
    const int t = threadIdx.x;                   // 64 threads
    float tv[TOPK]; int ti[TOPK];
    #pragma unroll
    for (int r = 0; r < TOPK; ++r) { tv[r] = -INFINITY; ti[r] = 0x7fffffff; }

    for (int j = t; j < n; j += 64) {
        const float v = s2[j];
        if (tk_better(v, j, tv[TOPK - 1], ti[TOPK - 1])) {
            int p = TOPK - 1;
            while (p > 0 && tk_better(v, j, tv[p - 1], ti[p - 1])) {
                tv[p] = tv[p - 1]; ti[p] = ti[p - 1]; --p;
            }
            tv[p] = v; ti[p] = j;
        }
    }
    #pragma unroll
    for (int r = 0; r < TOPK; ++r) { sv[t * TOPK + r] = tv[r]; si[t * TOPK + r] = ti[r]; }
    __syncthreads();

    if (t == 0) {
        float gv[TOPK]; int gi[TOPK];
        #pragma unroll
        for (int r = 0; r < TOPK; ++r) { gv[r] = -INFINITY; gi[r] = 0x7fffffff; }
        for (int c = 0; c < 64 * TOPK; ++c) {
            const float v = sv[c]; const int i = si[c];
            if (i == 0x7fffffff) continue;
            if (tk_better(v, i, gv[TOPK - 1], gi[TOPK - 1])) {
                int p = TOPK - 1;
                while (p > 0 && tk_better(v, i, gv[p - 1], gi[p - 1])) {
                    gv[p] = gv[p - 1]; gi[p] = gi[p - 1]; --p;
                }
                gv[p] = v; gi[p] = i;
            }
        }
        for (int r = 0; r < TOPK; ++r) { tv_out[r] = gv[r]; ti_out[r] = gi[r]; }
    }
}

// ---------------------------------------------------------------------------
// Kernel 3: per row i: vals_r = lrelu(s1[i] + tv[r]); softmax over 16;
// emit edge_index (src = shared indices, dst = i) and edge_weight.
// ---------------------------------------------------------------------------
__global__ void emit_kernel(const float* __restrict__ s1,
                            const float* __restrict__ tv,
                            const int* __restrict__ ti,
                            int* __restrict__ src,
                            int* __restrict__ dst,
                            float* __restrict__ wgt,
                            int n) {
    const int i = blockIdx.x * blockDim.x + threadIdx.x;
    if (i >= n) return;

    const float s = s1[i];
    float v[TOPK];
    #pragma unroll
    for (int r = 0; r < TOPK; ++r) {
        const float x = s + tv[r];
        v[r] = (x > 0.f) ? x : NEG_SLOPE * x;
    }
    // values sorted descending + monotone lrelu => v[0] is the max, but compute
    // max defensively.
    float mx = v[0];
    #pragma unroll
    for (int r = 1; r < TOPK; ++r) mx = fmaxf(mx, v[r]);
    float sum = 0.f;
    float e[TOPK];
    #pragma unroll
    for (int r = 0; r < TOPK; ++r) { e[r] = expf(v[r] - mx); sum += e[r]; }
    const float inv = 1.f / sum;
    #pragma unroll
    for (int r = 0; r < TOPK; ++r) {
        src[i * TOPK + r] = ti[r];
        dst[i * TOPK + r] = i;
        wgt[i * TOPK + r] = e[r] * inv;
    }
}

// ---------------------------------------------------------------------------
extern "C" void kernel_launch(void* const* d_in, const int* in_sizes, int n_in,
                              void* d_out, int out_size, void* d_ws, size_t ws_size,
                              hipStream_t stream) {
    const float* E = (const float*)d_in[0];   // (N, H)
    const float* W = (const float*)d_in[1];   // (H, H)
    const float* a = (const float*)d_in[2];   // (2H,)
    const int N = in_sizes[0] / H;            // 8192

    // Workspace layout (floats): B[128*16] | s1[N] | s2[N] | tv[16] | ti[16]
    float* ws  = (float*)d_ws;
    float* B   = ws;                  // 2048 floats
    float* s1  = B + H * 16;
    float* s2  = s1 + N;
    float* tv  = s2 + N;
    int*   ti  = (int*)(tv + TOPK);

    build_b_kernel<<<1, H, 0, stream>>>(W, a, B);
    gemv_wmma_kernel<<<N / 16, 32, 0, stream>>>(E, B, s1, s2);
    topk_kernel<<<1, 64, 0, stream>>>(s2, tv, ti, N);

    const int Nk = N * TOPK;
    int*   src = (int*)d_out;                 // edge_index row 0
    int*   dst = src + Nk;                    // edge_index row 1
    float* wgt = (float*)d_out + 2 * (size_t)Nk;  // edge_weight
    emit_kernel<<<(N + 255) / 256, 256, 0, stream>>>(s1, tv, ti, src, dst, wgt, N);
}